// DifferentiableCanny_44315472560276
// MI455X (gfx1250) — compile-verified
//
#include <hip/hip_runtime.h>
#include <stdint.h>

// Canny forward, fully fused, one pass over memory (32,3,512,512) fp32.
// Output: concat(high, low), each (32,1,512,512) fp32.
//
// Memory-bound: ~168 MB total traffic -> ~7.2us floor at 23.3 TB/s.
// gfx1250 path: B128 async global->LDS halo-tile copies (ASYNCcnt) + all
// intermediate stages in LDS + 128-bit non-temporal output stores.
//
// Column convention: local col c  <->  image col  tx0 - 4 + c   (4-aligned
// window so every 4-float quad is 16B aligned and fully in/out of image).
// Row origins per buffer: sRaw/sGray: ty0-3 ; sSmooth: ty0-2 ; sMag: ty0-1.

static constexpr int IMG   = 512;
static constexpr int TILE  = 32;
static constexpr int GROWS = 38;   // gray rows   (halo 3)
static constexpr int SROWS = 36;   // smooth rows (halo 2)
static constexpr int MROWS = 34;   // mag rows    (halo 1)
static constexpr int PITCH = 40;   // cols: image cols [tx0-4, tx0+36)

typedef float vfloat4 __attribute__((ext_vector_type(4)));

__global__ __launch_bounds__(256) void canny_fused_kernel(
    const float* __restrict__ x, float* __restrict__ out_hi, float* __restrict__ out_lo)
{
    __shared__ __align__(16) float sRaw[3][GROWS][PITCH]; // raw channels; [0] becomes gray
    __shared__ float sSmooth[SROWS][PITCH];
    __shared__ float sMag[MROWS][PITCH];

    const int tid = threadIdx.x;
    const int b   = blockIdx.z;
    const int tx0 = blockIdx.x * TILE;
    const int ty0 = blockIdx.y * TILE;

    const float* xb = x + (size_t)b * 3 * IMG * IMG;

    // ---- Stage 1: async B128 copies of the 38x40 halo window, 3 channels ----
    // 3 ch * 38 rows * 10 quads = 1140 B128 transfers per block.
    for (int j = tid; j < 3 * GROWS * 10; j += 256) {
        const int ch  = j / (GROWS * 10);
        const int rem = j - ch * (GROWS * 10);
        const int r   = rem / 10;
        const int q   = rem - r * 10;
        const int iy  = ty0 - 3 + r;
        const int ix  = tx0 - 4 + q * 4;     // 4-aligned: quad fully in or out
        if (((unsigned)iy < (unsigned)IMG) & ((unsigned)ix < (unsigned)IMG)) {
            const float* gp = xb + (size_t)ch * IMG * IMG + (size_t)iy * IMG + ix;
            uint64_t ga = (uint64_t)(uintptr_t)gp;
            uint32_t la = (uint32_t)(uintptr_t)&sRaw[ch][r][q * 4];
            asm volatile("global_load_async_to_lds_b128 %0, %1, off"
                         :: "v"(la), "v"(ga) : "memory");
        }
    }
    asm volatile("s_wait_asynccnt 0" ::: "memory");
    __syncthreads();

    // ---- Stage 2: grayscale mean, zero outside image (owner-exclusive r/w) ----
    for (int i = tid; i < GROWS * GROWS; i += 256) {
        const int r  = i / GROWS;
        const int c  = 1 + (i - r * GROWS);          // image cols tx0-3 .. tx0+34
        const int iy = ty0 - 3 + r;
        const int ix = tx0 - 4 + c;
        float g = 0.0f;
        if (((unsigned)iy < (unsigned)IMG) & ((unsigned)ix < (unsigned)IMG))
            g = (sRaw[0][r][c] + sRaw[1][r][c] + sRaw[2][r][c]) * (1.0f / 3.0f);
        sRaw[0][r][c] = g;
    }
    __syncthreads();

#define sGray (sRaw[0])

    // ---- Stage 3: 3x3 Gaussian (sigma=1), zero-padded like the reference ----
    const float GC = 0.2041799575f, GE = 0.1238414094f, GK = 0.0751136094f;
    for (int i = tid; i < SROWS * SROWS; i += 256) {
        const int r  = i / SROWS;
        const int c  = 2 + (i - r * SROWS);          // image cols tx0-2 .. tx0+33
        const int iy = ty0 - 2 + r;
        const int ix = tx0 - 4 + c;
        float v = 0.0f;
        if (((unsigned)iy < (unsigned)IMG) & ((unsigned)ix < (unsigned)IMG)) {
            const float a00 = sGray[r    ][c - 1], a01 = sGray[r    ][c], a02 = sGray[r    ][c + 1];
            const float a10 = sGray[r + 1][c - 1], a11 = sGray[r + 1][c], a12 = sGray[r + 1][c + 1];
            const float a20 = sGray[r + 2][c - 1], a21 = sGray[r + 2][c], a22 = sGray[r + 2][c + 1];
            v = GK * (a00 + a02 + a20 + a22) + GE * (a01 + a10 + a12 + a21) + GC * a11;
        }
        sSmooth[r][c] = v;
    }
    __syncthreads();

    // ---- Stage 4: Sobel magnitude, zero outside image (NMS zero-pad of mag) ----
    for (int i = tid; i < MROWS * MROWS; i += 256) {
        const int r  = i / MROWS;
        const int c  = 3 + (i - r * MROWS);          // image cols tx0-1 .. tx0+32
        const int iy = ty0 - 1 + r;
        const int ix = tx0 - 4 + c;
        float m = 0.0f;
        if (((unsigned)iy < (unsigned)IMG) & ((unsigned)ix < (unsigned)IMG)) {
            const float s00 = sSmooth[r    ][c - 1], s01 = sSmooth[r    ][c], s02 = sSmooth[r    ][c + 1];
            const float s10 = sSmooth[r + 1][c - 1],                           s12 = sSmooth[r + 1][c + 1];
            const float s20 = sSmooth[r + 2][c - 1], s21 = sSmooth[r + 2][c], s22 = sSmooth[r + 2][c + 1];
            const float gx = (s02 - s00) + 2.0f * (s12 - s10) + (s22 - s20);
            const float gy = (s20 - s00) + 2.0f * (s21 - s01) + (s22 - s02);
            m = sqrtf(gx * gx + gy * gy + 1e-6f);
        }
        sMag[r][c] = m;
    }
    __syncthreads();

    // ---- Stage 5: direction + NMS + soft thresholds; 4 pixels/thread, B128 NT stores ----
    const float T1 = 0.414213562f;   // tan(pi/8)
    const float T3 = 2.414213562f;   // tan(3pi/8)
    {
        const int t4 = tid * 4;                  // 256 threads * 4 = 1024 pixels
        const int py = t4 >> 5;                  // 0..31
        const int px0 = t4 & 31;                 // 0,4,...,28
        vfloat4 vhi, vlo;
#pragma unroll
        for (int k = 0; k < 4; ++k) {
            const int px = px0 + k;
            const int mr = py + 1, mc = px + 4;          // sMag coords of center
            const float m = sMag[mr][mc];

            // recompute gx,gy at center from sSmooth rows py+1..py+3, cols px+3..px+5
            const int sr = py + 1, sc = px + 3;
            const float s00 = sSmooth[sr    ][sc], s01 = sSmooth[sr    ][sc + 1], s02 = sSmooth[sr    ][sc + 2];
            const float s10 = sSmooth[sr + 1][sc],                                 s12 = sSmooth[sr + 1][sc + 2];
            const float s20 = sSmooth[sr + 2][sc], s21 = sSmooth[sr + 2][sc + 1], s22 = sSmooth[sr + 2][sc + 2];
            const float gx = (s02 - s00) + 2.0f * (s12 - s10) + (s22 - s20);
            const float gy = (s20 - s00) + 2.0f * (s21 - s01) + (s22 - s02);

            // bin = round(atan2(gy,gx)*4/pi) mod 4 via slope tests (mod-pi symmetric)
            const float ax = fabsf(gx), ay = fabsf(gy);
            int dy, dx;
            if (ay <= T1 * ax)        { dy = 0; dx = 1;  }   // bin 0: E/W
            else if (ay >= T3 * ax)   { dy = 1; dx = 0;  }   // bin 2: N/S
            else if (gx * gy > 0.0f)  { dy = 1; dx = 1;  }   // bin 1
            else                      { dy = 1; dx = -1; }   // bin 3

            const float np = sMag[mr + dy][mc + dx];
            const float nn = sMag[mr - dy][mc - dx];
            const float nms = ((m > np) && (m > nn)) ? m : 0.0f;

            vhi[k] = nms * (1.0f / (1.0f + __expf(-(nms - 0.3f) * 10.0f)));
            vlo[k] = nms * (1.0f / (1.0f + __expf(-(nms - 0.1f) * 10.0f)));
        }
        const size_t o = (size_t)b * IMG * IMG + (size_t)(ty0 + py) * IMG + (tx0 + px0);
        __builtin_nontemporal_store(vhi, (vfloat4*)&out_hi[o]);   // 16B-aligned, never re-read
        __builtin_nontemporal_store(vlo, (vfloat4*)&out_lo[o]);
    }
#undef sGray
}

extern "C" void kernel_launch(void* const* d_in, const int* in_sizes, int n_in,
                              void* d_out, int out_size, void* d_ws, size_t ws_size,
                              hipStream_t stream) {
    const float* x  = (const float*)d_in[0];
    float* out_hi   = (float*)d_out;
    float* out_lo   = out_hi + (size_t)32 * IMG * IMG;

    dim3 grid(IMG / TILE, IMG / TILE, 32);   // 16 x 16 x 32 tiles
    dim3 block(256);                         // 8 wave32s
    canny_fused_kernel<<<grid, block, 0, stream>>>(x, out_hi, out_lo);
}